// lstm_crf_13022340841483
// MI455X (gfx1250) — compile-verified
//
#include <hip/hip_runtime.h>
#include <math.h>

// ---------------- shapes ----------------
#define BB   64
#define TT   256
#define EE   300
#define HH   500
#define G4H  2000   // 4*H
#define D2H  1000   // 2*H
#define KT   64     // CRF tags
#define SOS  2
#define NEGV (-10000.0f)

typedef __attribute__((ext_vector_type(2))) float v2f;
typedef __attribute__((ext_vector_type(8))) float v8f;

__device__ __forceinline__ float sigf(float v) { return 1.0f / (1.0f + expf(-v)); }

// ---------------- f32 WMMA: D = A(16x4) * B(4x16) + C(16x16) ----------------
__device__ __forceinline__ v8f wmma4(v2f a, v2f b, v8f c) {
  // 8 args: (neg_a, A, neg_b, B, c_mod, C, reuse_a, reuse_b)
  return __builtin_amdgcn_wmma_f32_16x16x4_f32(false, a, false, b, (short)0, c,
                                               false, false);
}

// Accumulate a K-segment over a full 64-row strip (4 M-tiles) against one
// 16-col weight tile. Weight fragment loaded once per k-step, reused 4x.
// Arow*/Brow already offset by (row*ld + kb); lane mapping per ISA 7.12.2.
__device__ __forceinline__ void gemm_strip(const float* __restrict__ a0, int lda,
                                           const float* __restrict__ bw,
                                           int Kd,
                                           v8f& acc0, v8f& acc1, v8f& acc2, v8f& acc3) {
  const float* a1 = a0 + (size_t)16 * lda;
  const float* a2 = a0 + (size_t)32 * lda;
  const float* a3 = a0 + (size_t)48 * lda;
#pragma unroll 2
  for (int k = 0; k < Kd; k += 4) {
    v2f bf  = *(const v2f*)(bw + k);
    v2f af0 = *(const v2f*)(a0 + k);
    v2f af1 = *(const v2f*)(a1 + k);
    v2f af2 = *(const v2f*)(a2 + k);
    v2f af3 = *(const v2f*)(a3 + k);
    acc0 = wmma4(af0, bf, acc0);
    acc1 = wmma4(af1, bf, acc1);
    acc2 = wmma4(af2, bf, acc2);
    acc3 = wmma4(af3, bf, acc3);
  }
}

__device__ __forceinline__ void store_tile(float* __restrict__ z, int ldz,
                                           int m0, int n, int half,
                                           v8f acc, float bv) {
#pragma unroll
  for (int i = 0; i < 8; ++i)
    z[(size_t)(m0 + i + 8 * half) * ldz + n] = acc[i] + bv;
}

// ---------------- utility kernels ----------------
__global__ void zero_kernel(float* __restrict__ p, int n) {
  int i = blockIdx.x * blockDim.x + threadIdx.x;
  if (i < n) p[i] = 0.0f;
}

__global__ void embed_kernel(const int* __restrict__ x,
                             const float* __restrict__ emb,
                             float* __restrict__ xs) {
  long long idx = (long long)blockIdx.x * blockDim.x + threadIdx.x;
  const long long N = (long long)TT * BB * EE;
  if (idx >= N) return;
  int e  = (int)(idx % EE);
  int tb = (int)(idx / EE);
  int b  = tb % BB;
  int t  = tb / BB;
  xs[idx] = emb[(size_t)x[b * TT + t] * EE + e];
}

// ---------------- LSTM step GEMM: z = [xt,h] @ [Wih,Whh]^T + b ----------------
// One wave owns a full 64x16 column strip of z[B,4H] (4 M-tiles, shared B frag).
// gridDim.y selects direction (0=fwd, 1=bwd).
__global__ void lstm_gemm_dual(
    const float* __restrict__ XtF, const float* __restrict__ XtB,
    int ldx, int Din,
    const float* __restrict__ WihF, const float* __restrict__ WhhF,
    const float* __restrict__ bF,
    const float* __restrict__ WihB, const float* __restrict__ WhhB,
    const float* __restrict__ bB,
    const float* __restrict__ hF, const float* __restrict__ hB,
    float* __restrict__ zF, float* __restrict__ zB) {
  const int wid    = blockIdx.x * (blockDim.x >> 5) + (threadIdx.x >> 5);
  const int tilesN = G4H / 16;  // 125 column strips
  if (wid >= tilesN) return;

  const int dir = blockIdx.y;
  const float* Xt  = dir ? XtB : XtF;
  const float* Wih = dir ? WihB : WihF;
  const float* Whh = dir ? WhhB : WhhF;
  const float* bia = dir ? bB : bF;
  const float* hp  = dir ? hB : hF;
  float*       z   = dir ? zB : zF;

  const int n0   = wid * 16;
  const int lane = threadIdx.x & 31;
  const int r    = lane & 15;
  const int half = lane >> 4;
  const int kb   = half << 1;

  v8f acc0 = {}, acc1 = {}, acc2 = {}, acc3 = {};

  // input-projection segment (K = Din)
  gemm_strip(Xt + (size_t)r * ldx + kb, ldx,
             Wih + (size_t)(n0 + r) * Din + kb, Din,
             acc0, acc1, acc2, acc3);
  // recurrent segment (K = H)
  gemm_strip(hp + (size_t)r * HH + kb, HH,
             Whh + (size_t)(n0 + r) * HH + kb, HH,
             acc0, acc1, acc2, acc3);

  const float bv = bia[n0 + r];
  const int   n  = n0 + r;
  store_tile(z, G4H, 0,  n, half, acc0, bv);
  store_tile(z, G4H, 16, n, half, acc1, bv);
  store_tile(z, G4H, 32, n, half, acc2, bv);
  store_tile(z, G4H, 48, n, half, acc3, bv);
}

// ---------------- LSTM gate/elementwise update ----------------
__global__ void lstm_gate_dual(
    const float* __restrict__ zF, const float* __restrict__ zB,
    float* __restrict__ hF, float* __restrict__ cF,
    float* __restrict__ hB, float* __restrict__ cB,
    float* __restrict__ outF, float* __restrict__ outB) {
  int idx = blockIdx.x * blockDim.x + threadIdx.x;
  if (idx >= BB * HH) return;
  const int dir  = blockIdx.y;
  const float* z = dir ? zB : zF;
  float* h = dir ? hB : hF;
  float* c = dir ? cB : cF;
  float* o = dir ? outB : outF;

  int b = idx / HH;
  int j = idx - b * HH;
  const float* zr = z + (size_t)b * G4H;
  float ig = sigf(zr[j]);
  float fg = sigf(zr[HH + j]);
  float gg = tanhf(zr[2 * HH + j]);
  float og = sigf(zr[3 * HH + j]);
  float cn = fg * c[idx] + ig * gg;
  float hn = og * tanhf(cn);
  c[idx] = cn;
  h[idx] = hn;
  o[(size_t)b * D2H + j] = hn;  // into [T,B,2H] concat buffer (offset pre-applied)
}

// ---------------- emit = h1 @ Wout^T + bout, masked by (x>0) ----------------
// One wave owns a 64x16 strip (4 M-tiles sharing the Wout fragment).
__global__ void out_gemm_mask(const float* __restrict__ X,   // [T*B, 2H]
                              const float* __restrict__ W,   // [K, 2H]
                              const float* __restrict__ bias,
                              const int* __restrict__ tok,   // x [B,T]
                              float* __restrict__ Y) {       // [T*B, K]
  const int wid    = blockIdx.x * (blockDim.x >> 5) + (threadIdx.x >> 5);
  const int tilesN = KT / 16;        // 4
  const int M      = TT * BB;        // 16384
  const int strips = M / 64;         // 256
  if (wid >= strips * tilesN) return;

  const int m0   = (wid / tilesN) * 64;
  const int n0   = (wid % tilesN) * 16;
  const int lane = threadIdx.x & 31;
  const int r    = lane & 15;
  const int half = lane >> 4;
  const int kb   = half << 1;

  v8f acc0 = {}, acc1 = {}, acc2 = {}, acc3 = {};
  gemm_strip(X + (size_t)(m0 + r) * D2H + kb, D2H,
             W + (size_t)(n0 + r) * D2H + kb, D2H,
             acc0, acc1, acc2, acc3);

  const float bv = bias[n0 + r];
  const int   n  = n0 + r;
  v8f accs[4] = {acc0, acc1, acc2, acc3};
#pragma unroll
  for (int mtile = 0; mtile < 4; ++mtile) {
#pragma unroll
    for (int i = 0; i < 8; ++i) {
      int row = m0 + mtile * 16 + i + 8 * half;
      int t = row / BB;
      int b = row - t * BB;
      float mk = (tok[b * TT + t] > 0) ? 1.0f : 0.0f;
      Y[(size_t)row * KT + n] = (accs[mtile][i] + bv) * mk;
    }
  }
}

// ---------------- CRF forward + gold path; one block per batch row ----------------
__global__ void crf_kernel(const float* __restrict__ emit,  // [T,B,K]
                           const int* __restrict__ x,       // [B,T]
                           const int* __restrict__ y0,      // [B,T]
                           const float* __restrict__ trans, // [K,K]
                           float* __restrict__ out) {       // [B]
  __shared__ float tr[KT * KT];
  __shared__ float score[KT];
  const int b = blockIdx.x;
  const int k = threadIdx.x;

  for (int j = 0; j < KT; ++j) tr[k * KT + j] = trans[k * KT + j];
  score[k] = (k == SOS) ? 0.0f : NEGV;
  __syncthreads();

  float gold = 0.0f;
  int prev = SOS;
  for (int t = 0; t < TT; ++t) {
    const float m = (x[b * TT + t] > 0) ? 1.0f : 0.0f;
    const float* et = emit + ((size_t)t * BB + b) * KT;
    // new[k] = logsumexp_j(score[j] + trans[k][j]) + e[k]
    float mx = -3.0e38f;
#pragma unroll
    for (int j = 0; j < KT; ++j) mx = fmaxf(mx, score[j] + tr[k * KT + j]);
    float s = 0.0f;
#pragma unroll
    for (int j = 0; j < KT; ++j) s += expf(score[j] + tr[k * KT + j] - mx);
    float nv  = mx + logf(s) + et[k];
    float cur = score[k];
    __syncthreads();                       // all reads done
    score[k] = (m > 0.0f) ? nv : cur;
    if (k == 0) {                          // gold path (serial, cheap)
      int tag = y0[b * TT + t];
      gold += et[tag] + tr[tag * KT + prev] * m;
      prev = tag;
    }
    __syncthreads();
  }
  if (k == 0) {
    float mx = -3.0e38f;
    for (int j = 0; j < KT; ++j) mx = fmaxf(mx, score[j]);
    float s = 0.0f;
    for (int j = 0; j < KT; ++j) s += expf(score[j] - mx);
    out[b] = mx + logf(s) - gold;
  }
}

// ---------------- host orchestration ----------------
extern "C" void kernel_launch(void* const* d_in, const int* in_sizes, int n_in,
                              void* d_out, int out_size, void* d_ws, size_t ws_size,
                              hipStream_t stream) {
  (void)in_sizes; (void)n_in; (void)out_size; (void)ws_size;

  const int*   x     = (const int*)d_in[0];
  const int*   y0    = (const int*)d_in[1];
  const float* embed = (const float*)d_in[2];
  const float* Wih0f = (const float*)d_in[3];
  const float* Whh0f = (const float*)d_in[4];
  const float* b0f   = (const float*)d_in[5];
  const float* Wih0b = (const float*)d_in[6];
  const float* Whh0b = (const float*)d_in[7];
  const float* b0b   = (const float*)d_in[8];
  const float* Wih1f = (const float*)d_in[9];
  const float* Whh1f = (const float*)d_in[10];
  const float* b1f   = (const float*)d_in[11];
  const float* Wih1b = (const float*)d_in[12];
  const float* Whh1b = (const float*)d_in[13];
  const float* b1b   = (const float*)d_in[14];
  const float* Wout  = (const float*)d_in[15];
  const float* bout  = (const float*)d_in[16];
  const float* trans = (const float*)d_in[17];
  float* out = (float*)d_out;

  // workspace layout (floats); total ~39.2M floats (~157 MB)
  float* ws = (float*)d_ws;
  size_t off = 0;
  float* xs   = ws + off; off += (size_t)TT * BB * EE;    // [T,B,E]
  float* h0c  = ws + off; off += (size_t)TT * BB * D2H;   // [T,B,2H]
  float* h1c  = ws + off; off += (size_t)TT * BB * D2H;   // [T,B,2H]
  float* emit = ws + off; off += (size_t)TT * BB * KT;    // [T,B,K]
  float* zf   = ws + off; off += (size_t)BB * G4H;
  float* zb   = ws + off; off += (size_t)BB * G4H;
  float* hf   = ws + off; off += (size_t)BB * HH;
  float* cf   = ws + off; off += (size_t)BB * HH;
  float* hb   = ws + off; off += (size_t)BB * HH;
  float* cb   = ws + off; off += (size_t)BB * HH;

  const int nState = 4 * BB * HH;  // hf,cf,hb,cb contiguous

  // 1) embedding gather
  {
    long long n = (long long)TT * BB * EE;
    embed_kernel<<<(unsigned)((n + 255) / 256), 256, 0, stream>>>(x, embed, xs);
  }

  const dim3 gridGemm(16, 2, 1);   // 16 blocks x 8 waves = 128 >= 125 strips, x2 dirs
  const dim3 gridGate((BB * HH + 255) / 256, 2, 1);

  // 2) layer 0 recurrence (fwd + bwd concurrently)
  zero_kernel<<<(nState + 255) / 256, 256, 0, stream>>>(hf, nState);
  for (int s = 0; s < TT; ++s) {
    int tF = s, tB = TT - 1 - s;
    lstm_gemm_dual<<<gridGemm, 256, 0, stream>>>(
        xs + (size_t)tF * BB * EE, xs + (size_t)tB * BB * EE, EE, EE,
        Wih0f, Whh0f, b0f, Wih0b, Whh0b, b0b, hf, hb, zf, zb);
    lstm_gate_dual<<<gridGate, 256, 0, stream>>>(
        zf, zb, hf, cf, hb, cb,
        h0c + (size_t)tF * BB * D2H,            // fwd -> cols [0,H)
        h0c + (size_t)tB * BB * D2H + HH);      // bwd -> cols [H,2H)
  }

  // 3) layer 1 recurrence
  zero_kernel<<<(nState + 255) / 256, 256, 0, stream>>>(hf, nState);
  for (int s = 0; s < TT; ++s) {
    int tF = s, tB = TT - 1 - s;
    lstm_gemm_dual<<<gridGemm, 256, 0, stream>>>(
        h0c + (size_t)tF * BB * D2H, h0c + (size_t)tB * BB * D2H, D2H, D2H,
        Wih1f, Whh1f, b1f, Wih1b, Whh1b, b1b, hf, hb, zf, zb);
    lstm_gate_dual<<<gridGate, 256, 0, stream>>>(
        zf, zb, hf, cf, hb, cb,
        h1c + (size_t)tF * BB * D2H,
        h1c + (size_t)tB * BB * D2H + HH);
  }

  // 4) emit projection + mask (256 strips x 4 N-tiles = 1024 waves -> 128 blocks)
  out_gemm_mask<<<128, 256, 0, stream>>>(h1c, Wout, bout, x, emit);

  // 5) CRF forward + gold, one block per batch row
  crf_kernel<<<BB, KT, 0, stream>>>(emit, x, y0, trans, out);
}